// Model_32212254720224
// MI455X (gfx1250) — compile-verified
//
#include <hip/hip_runtime.h>
#include <hip/hip_bf16.h>

// ---------------------------------------------------------------------------
// Ragged paged-KV index gather (SGLang create_kv_indices style), MI455X/gfx1250.
//
// out[t] = req_to_token[req_pool_indices[r] * MAXLEN + kv_start_idx[r] + (t - kv_indptr[r])]
// where r = searchsorted_right(kv_indptr, t) - 1.
//
// Memory-bound gather: no FLOPs -> WMMA not applicable. Optimization targets:
// coalesced unit-stride copies, one block-level search instead of per-element,
// CDNA5 async global->LDS staging of the descriptor tables (ASYNCcnt path).
// ---------------------------------------------------------------------------

#define KV_B       64          // number of requests (kv_indptr has KV_B+1 entries)
#define KV_MAXLEN  16384       // row stride of req_to_token
#define THREADS    256         // 8 wave32s per block
#define EPT        2           // elements per thread
#define TILE       (THREADS * EPT)   // 512 <= min request len (1024) -> tile spans <=2 requests

// 193 staged dwords: kv_indptr[65] | req_pool_indices[64] | kv_start_idx[64]
#define N_INDPTR   (KV_B + 1)
#define N_TAB      (N_INDPTR + KV_B + KV_B)

#if defined(__gfx1250__) && __has_builtin(__builtin_amdgcn_global_load_async_to_lds_b32)
#define USE_ASYNC_LDS 1
#else
#define USE_ASYNC_LDS 0
#endif

#if USE_ASYNC_LDS
typedef __attribute__((address_space(1))) int as1_int;   // global
typedef __attribute__((address_space(3))) int as3_int;   // LDS
#endif

__global__ __launch_bounds__(THREADS) void Model_32212254720224_kernel(
    const int* __restrict__ req_to_token,      // [POOL * MAXLEN]
    const int* __restrict__ req_pool_indices,  // [B]
    const int* __restrict__ kv_indptr,         // [B+1]
    const int* __restrict__ kv_start_idx,      // [B]
    float*     __restrict__ out,               // [total]
    int total)
{
    __shared__ int s_tab[N_TAB];
    const int* s_indptr = s_tab;                   // 65
    const int* s_rows   = s_tab + N_INDPTR;        // 64
    const int* s_start  = s_tab + N_INDPTR + KV_B; // 64

    const int tid = threadIdx.x;

    // ---- Stage descriptor tables into LDS -------------------------------
    if (tid < N_TAB) {
        const int* src = (tid < N_INDPTR)
                             ? (kv_indptr + tid)
                             : (tid < N_INDPTR + KV_B)
                                   ? (req_pool_indices + (tid - N_INDPTR))
                                   : (kv_start_idx + (tid - N_INDPTR - KV_B));
#if USE_ASYNC_LDS
        // CDNA5 async copy: global -> LDS without touching VGPR data,
        // tracked by ASYNCcnt (s_wait_asynccnt), not LOADcnt/DScnt.
        __builtin_amdgcn_global_load_async_to_lds_b32(
            (as1_int*)const_cast<int*>(src),
            (as3_int*)&s_tab[tid],
            /*offset=*/0, /*cpol=*/0);
#else
        s_tab[tid] = *src;
#endif
    }
#if USE_ASYNC_LDS
#if __has_builtin(__builtin_amdgcn_s_wait_asynccnt)
    __builtin_amdgcn_s_wait_asynccnt(0);
#else
    asm volatile("s_wait_asynccnt 0" ::: "memory");
#endif
#endif
    __syncthreads();

    // ---- One branchless searchsorted per block --------------------------
    // r0 = largest r in [0,63] with kv_indptr[r] <= blockStart.
    const int blockStart = blockIdx.x * TILE;
    int r0 = 0;
#pragma unroll
    for (int step = 32; step >= 1; step >>= 1) {
        int cand = r0 + step;
        if (cand <= KV_B - 1 && s_indptr[cand] <= blockStart) r0 = cand;
    }

    // ---- Copy: contiguous per-request source segments -------------------
    // Tile (512) <= min request length (1024) -> each element advances r by
    // at most 1 past r0; the while loop is robust regardless.
#pragma unroll
    for (int k = 0; k < EPT; ++k) {
        const int t = blockStart + k * THREADS + tid;
        if (t < total) {
            int r = r0;
            while (t >= s_indptr[r + 1]) ++r;          // bounded (<=1 iter here)
            const int off = t - s_indptr[r];
            const long long row = (long long)s_rows[r];
            const int col = s_start[r] + off;
            const int tok = req_to_token[row * (long long)KV_MAXLEN + col];
            out[t] = (float)tok;                       // tok < 2^24: exact in f32
        }
    }
}

extern "C" void kernel_launch(void* const* d_in, const int* in_sizes, int n_in,
                              void* d_out, int out_size, void* d_ws, size_t ws_size,
                              hipStream_t stream) {
    // setup_inputs() order:
    //   0: req_to_token [POOL*MAXLEN]  1: req_pool_indices [B]
    //   2: page_kernel_lens [B] (unused)  3: kv_indptr [B+1]  4: kv_start_idx [B]
    const int* req_to_token     = (const int*)d_in[0];
    const int* req_pool_indices = (const int*)d_in[1];
    const int* kv_indptr        = (const int*)d_in[3];
    const int* kv_start_idx     = (const int*)d_in[4];
    float* out = (float*)d_out;

    const int total = out_size;                 // == kv_indptr[-1] by construction
    const int grid  = (total + TILE - 1) / TILE;

    Model_32212254720224_kernel<<<grid, THREADS, 0, stream>>>(
        req_to_token, req_pool_indices, kv_indptr, kv_start_idx, out, total);
}